// LongTermEmbedding_18957985645139
// MI455X (gfx1250) — compile-verified
//
#include <hip/hip_runtime.h>

typedef __attribute__((ext_vector_type(2))) float v2f;
typedef __attribute__((ext_vector_type(8))) float v8f;

#define DAYS_PER_BLOCK 16
#define BLOCK_THREADS  256
#define WAVES_PER_BLK  (BLOCK_THREADS / 32)
#define D_NEWS         64
#define D_CAT          16
#define D_TOT          80
#define PART_STRIDE    81   // pad 80 -> 81 words: lane halves hit different LDS banks

__global__ __launch_bounds__(BLOCK_THREADS)
void longterm_embed_segmean_wmma(const int* __restrict__ news_ids,
                                 const int* __restrict__ cat_ids,
                                 const int* __restrict__ day_ids,
                                 const float* __restrict__ delta_days,
                                 const float* __restrict__ news_table,
                                 const float* __restrict__ cat_table,
                                 float* __restrict__ out,
                                 int T, int M) {
    __shared__ int   s_bounds[DAYS_PER_BLOCK + 1];
    __shared__ float s_part[WAVES_PER_BLK][DAYS_PER_BLOCK * PART_STRIDE];

    const int tid      = threadIdx.x;
    const int day_base = blockIdx.x * DAYS_PER_BLOCK;

    // --- 17 lower-bound binary searches give segment bounds AND counts ---
    if (tid <= DAYS_PER_BLOCK) {
        int target = day_base + tid;
        int lo = 0, hi = T;
        while (lo < hi) {
            int mid = (lo + hi) >> 1;
            if (day_ids[mid] < target) lo = mid + 1; else hi = mid;
        }
        s_bounds[tid] = lo;
    }
    __syncthreads();

    const int seg_lo = s_bounds[0];
    const int seg_hi = s_bounds[DAYS_PER_BLOCK];
    const int len    = seg_hi - seg_lo;

    const int wave = tid >> 5;        // wave32 on gfx1250
    const int lane = tid & 31;
    const int half = lane >> 4;       // 0: lanes 0-15, 1: lanes 16-31
    const int kb   = half * 2;        // K base for A/B per documented VGPR layout
    const int nrow = lane & 15;       // M-row (for A) / N-col (for B)

    // Per-lane segment bounds: day(i) == day_base+nrow  <=>  b_lo <= i < b_hi.
    // Replaces the day_ids gather in the inner loop with two integer compares.
    const int b_lo = s_bounds[nrow];
    const int b_hi = s_bounds[nrow + 1];

    const int per_wave = (len + WAVES_PER_BLK - 1) / WAVES_PER_BLK;
    const int wstart   = seg_lo + wave * per_wave;
    const int wend     = min(wstart + per_wave, seg_hi);

    v8f acc[5] = {v8f{}, v8f{}, v8f{}, v8f{}, v8f{}};

    #pragma unroll 2
    for (int t0 = wstart; t0 < wend; t0 += 4) {
        const int i0 = t0 + kb;
        const int i1 = i0 + 1;
        const int c0 = min(i0, T - 1);   // clamped gather index; indicator is 0 past wend
        const int c1 = min(i1, T - 1);

        // A selection matrix via register-resident bounds (branch-free)
        v2f a;
        a.x = (i0 < wend && i0 >= b_lo && i0 < b_hi) ? 1.0f : 0.0f;
        a.y = (i1 < wend && i1 >= b_lo && i1 < b_hi) ? 1.0f : 0.0f;

        // 32-bit flat element offsets -> saddr + voffset (scale_offset) loads
        const int n0 = news_ids[c0] * D_NEWS;
        const int n1 = news_ids[c1] * D_NEWS;
        const int g0 = cat_ids[c0]  * D_CAT;
        const int g1 = cat_ids[c1]  * D_CAT;

        // chunks 0..3: news dims (16 lanes read 64 consecutive bytes per row)
        #pragma unroll
        for (int c = 0; c < 4; ++c) {
            v2f bm;
            bm.x = news_table[n0 + c * 16 + nrow];
            bm.y = news_table[n1 + c * 16 + nrow];
            acc[c] = __builtin_amdgcn_wmma_f32_16x16x4_f32(
                false, a, false, bm, (short)0, acc[c], false, false);
        }
        // chunk 4: category dims
        {
            v2f bm;
            bm.x = cat_table[g0 + nrow];
            bm.y = cat_table[g1 + nrow];
            acc[4] = __builtin_amdgcn_wmma_f32_16x16x4_f32(
                false, a, false, bm, (short)0, acc[4], false, false);
        }
    }

    // --- spill per-wave partial D tiles to LDS (documented C/D layout) ---
    float* pw = &s_part[wave][0];
    const int mbase = half * 8;       // lanes 16-31 hold rows M=8..15
    #pragma unroll
    for (int c = 0; c < 5; ++c) {
        #pragma unroll
        for (int i = 0; i < 8; ++i) {
            pw[(mbase + i) * PART_STRIDE + c * 16 + nrow] = acc[c][i];
        }
    }
    __syncthreads();

    // --- cross-wave reduce, divide by count, store Z ---
    for (int j = tid; j < DAYS_PER_BLOCK * D_TOT; j += BLOCK_THREADS) {
        const int m = j / D_TOT;
        const int d = j - m * D_TOT;
        float s = 0.0f;
        #pragma unroll
        for (int w = 0; w < WAVES_PER_BLK; ++w)
            s += s_part[w][m * PART_STRIDE + d];
        const float cnt = (float)(s_bounds[m + 1] - s_bounds[m]);
        out[(long long)(day_base + m) * D_TOT + d] = s / fmaxf(cnt, 1.0f);
    }

    // --- delta_t passthrough appended after Z ---
    if (tid < DAYS_PER_BLOCK && (day_base + tid) < M) {
        out[(long long)M * D_TOT + day_base + tid] = delta_days[day_base + tid];
    }
}

extern "C" void kernel_launch(void* const* d_in, const int* in_sizes, int n_in,
                              void* d_out, int out_size, void* d_ws, size_t ws_size,
                              hipStream_t stream) {
    const int*   news_ids   = (const int*)d_in[0];
    const int*   cat_ids    = (const int*)d_in[1];
    const int*   day_ids    = (const int*)d_in[2];
    const float* delta_days = (const float*)d_in[3];
    const float* news_table = (const float*)d_in[4];
    const float* cat_table  = (const float*)d_in[5];
    float*       out        = (float*)d_out;

    const int T = in_sizes[0];
    const int M = in_sizes[3];

    const int blocks = (M + DAYS_PER_BLOCK - 1) / DAYS_PER_BLOCK;  // 512 for M=8192
    longterm_embed_segmean_wmma<<<blocks, BLOCK_THREADS, 0, stream>>>(
        news_ids, cat_ids, day_ids, delta_days, news_table, cat_table, out, T, M);
}